// MultiFrameWienerFilter_16870631538826
// MI455X (gfx1250) — compile-verified
//
#include <hip/hip_runtime.h>
#include <stdint.h>

// Problem constants from the reference
#define PB 4
#define PT 600
#define PF 481
#define PN 5
#define LOOKA 1

constexpr int NPROB = PB * PT * PF;      // 1,154,400 independent 5x5 complex solves
constexpr int BLK   = 256;
constexpr int RB    = PN * PN * 2 * 4;   // 200 bytes of iRxx per problem
constexpr int VB    = PN * 2 * 4;        // 40  bytes of ifc  per problem

typedef float v2f __attribute__((ext_vector_type(2)));

__device__ __forceinline__ void cswap(bool c, float& a, float& b) {
    float t = c ? b : a;
    b = c ? a : b;
    a = t;
}

// One 16-byte async global->LDS lane transfer (CDNA5 ASYNCcnt path).
// VDST = LDS byte address, VADDR = 32-bit offset, SADDR = 64-bit base.
// Non-temporal: iRxx/ifc are streamed once (231+46 MB > 192 MB L2) —
// don't let them evict the reusable spec window.
__device__ __forceinline__ void async_ld16_nt(uint32_t lds_addr, uint32_t goff, const void* sbase) {
    asm volatile("global_load_async_to_lds_b128 %0, %1, %2 th:TH_LOAD_NT"
                 :: "v"(lds_addr), "v"(goff), "s"(sbase)
                 : "memory");
}

__global__ __launch_bounds__(BLK)
void mfwf_kernel(const float* __restrict__ spec,   // (B,1,T,F,2)
                 const float* __restrict__ ifc,    // (B,T,F,10)
                 const float* __restrict__ iRxx,   // (B,T,F,50)
                 float* __restrict__ out)          // (B,1,T,F,2)
{
    __shared__ float sR[BLK * 50];   // 51200 B
    __shared__ float sV[BLK * 10];   // 10240 B

    const int tid        = threadIdx.x;
    const int blockStart = blockIdx.x * BLK;
    const int nprob      = min(BLK, NPROB - blockStart);

    // ---- stage this block's contiguous iRxx / ifc slabs into LDS (async DMA) ----
    {
        const uint32_t bytesR = (uint32_t)nprob * RB;           // multiple of 16
        const char*    gR     = (const char*)iRxx + (size_t)blockStart * RB;
        const uint32_t ldsR   = (uint32_t)(uintptr_t)(&sR[0]);
        for (uint32_t off = (uint32_t)tid * 16u; off < bytesR; off += BLK * 16u)
            async_ld16_nt(ldsR + off, off, gR);

        const uint32_t bytesV = (uint32_t)nprob * VB;           // multiple of 16
        const char*    gV     = (const char*)ifc + (size_t)blockStart * VB;
        const uint32_t ldsV   = (uint32_t)(uintptr_t)(&sV[0]);
        for (uint32_t off = (uint32_t)tid * 16u; off < bytesV; off += BLK * 16u)
            async_ld16_nt(ldsV + off, off, gV);

        asm volatile("s_wait_asynccnt 0" ::: "memory");
    }
    __syncthreads();

    if (tid < nprob) {
        const int p  = blockStart + tid;
        const int f  = p % PF;
        const int bt = p / PF;
        const int t  = bt % PT;
        const int b  = bt / PT;

        const float* rp = &sR[tid * 50];   // stride-50 floats: bank-conflict free
        const float* vp = &sV[tid * 10];

        // Load 5x5 complex matrix
        float Rr[5][5], Ri[5][5];
#pragma unroll
        for (int i = 0; i < 5; ++i)
#pragma unroll
            for (int j = 0; j < 5; ++j) {
                Rr[i][j] = rp[(i * 5 + j) * 2];
                Ri[i][j] = rp[(i * 5 + j) * 2 + 1];
            }

        // Hermitianize: upper (j>i) = conj(lower), diagonal forced real
#pragma unroll
        for (int i = 0; i < 5; ++i) {
            Ri[i][i] = 0.0f;
#pragma unroll
            for (int j = i + 1; j < 5; ++j) {
                Rr[i][j] =  Rr[j][i];
                Ri[i][j] = -Ri[j][i];
            }
        }

        // Diagonal loading
        const float tr = Rr[0][0] + Rr[1][1] + Rr[2][2] + Rr[3][3] + Rr[4][4];
        const float dl = tr * 1e-7f + 1e-8f;
#pragma unroll
        for (int i = 0; i < 5; ++i) Rr[i][i] += dl;

        // RHS
        float br[5], bi[5];
#pragma unroll
        for (int n = 0; n < 5; ++n) {
            br[n] = vp[2 * n];
            bi[n] = vp[2 * n + 1];
        }

        // Gaussian elimination with partial pivoting on |re|+|im| (branchless swaps)
#pragma unroll
        for (int k = 0; k < 5; ++k) {
#pragma unroll
            for (int r = k + 1; r < 5; ++r) {
                const bool sw = (fabsf(Rr[r][k]) + fabsf(Ri[r][k])) >
                                (fabsf(Rr[k][k]) + fabsf(Ri[k][k]));
#pragma unroll
                for (int j = k; j < 5; ++j) {
                    cswap(sw, Rr[k][j], Rr[r][j]);
                    cswap(sw, Ri[k][j], Ri[r][j]);
                }
                cswap(sw, br[k], br[r]);
                cswap(sw, bi[k], bi[r]);
            }
            const float pr  = Rr[k][k], pi = Ri[k][k];
            const float inv = 1.0f / (pr * pr + pi * pi);
            const float ipr =  pr * inv;
            const float ipi = -pi * inv;
#pragma unroll
            for (int i = k + 1; i < 5; ++i) {
                const float ar = Rr[i][k], ai = Ri[i][k];
                const float fr = ar * ipr - ai * ipi;
                const float fi = ar * ipi + ai * ipr;
#pragma unroll
                for (int j = k + 1; j < 5; ++j) {
                    Rr[i][j] -= fr * Rr[k][j] - fi * Ri[k][j];
                    Ri[i][j] -= fr * Ri[k][j] + fi * Rr[k][j];
                }
                br[i] -= fr * br[k] - fi * bi[k];
                bi[i] -= fr * bi[k] + fi * br[k];
            }
        }

        // Back substitution
        float wr[5], wi[5];
#pragma unroll
        for (int k = 4; k >= 0; --k) {
            float sr = br[k], si = bi[k];
#pragma unroll
            for (int j = k + 1; j < 5; ++j) {
                sr -= Rr[k][j] * wr[j] - Ri[k][j] * wi[j];
                si -= Rr[k][j] * wi[j] + Ri[k][j] * wr[j];
            }
            const float pr  = Rr[k][k], pi = Ri[k][k];
            const float inv = 1.0f / (pr * pr + pi * pi);
            wr[k] = (sr * pr + si * pi) * inv;
            wi[k] = (si * pr - sr * pi) * inv;
        }

        // Windowed complex dot: out = sum_n spec[t+n-3] * w[n]  (zero-padded in t)
        // spec stays at default (RT) temporal hint: 9 MB with 5x reuse -> keep in L2.
        float outr = 0.0f, outi = 0.0f;
        const size_t fstride2 = (size_t)PF * 2;
        const float* sb = spec + (size_t)b * PT * fstride2 + (size_t)f * 2;
#pragma unroll
        for (int n = 0; n < 5; ++n) {
            const int tt = t + n - (PN - 1 - LOOKA);
            if (tt >= 0 && tt < PT) {
                const float sr2 = sb[(size_t)tt * fstride2];
                const float si2 = sb[(size_t)tt * fstride2 + 1];
                outr += sr2 * wr[n] - si2 * wi[n];
                outi += sr2 * wi[n] + si2 * wr[n];
            }
        }

        // Output is written once, never read back -> non-temporal 64-bit store.
        v2f ov;
        ov.x = outr;
        ov.y = outi;
        __builtin_nontemporal_store(ov, (v2f*)(out + (size_t)p * 2));
    }
}

extern "C" void kernel_launch(void* const* d_in, const int* in_sizes, int n_in,
                              void* d_out, int out_size, void* d_ws, size_t ws_size,
                              hipStream_t stream) {
    const float* spec = (const float*)d_in[0];
    const float* ifc  = (const float*)d_in[1];
    const float* iRxx = (const float*)d_in[2];
    float*       out  = (float*)d_out;

    const int nblocks = (NPROB + BLK - 1) / BLK;
    hipLaunchKernelGGL(mfwf_kernel, dim3(nblocks), dim3(BLK), 0, stream,
                       spec, ifc, iRxx, out);
}